// Sampler_64733747085712
// MI455X (gfx1250) — compile-verified
//
#include <hip/hip_runtime.h>
#include <hip/hip_bf16.h>
#include <stdint.h>

#define B_ROWS   256
#define VOCAB    128000
#define VOCAB4   (VOCAB / 4)
#define TOPK     256
#define IGNORED_V (-3000.0f)
#define NBINS    16384          // 14-bit key prefix
#define BSHIFT   18             // 32 - 14
#define SEG      (NBINS / 256)  // bins per scan-thread segment = 64
#define THREADS  512            // 16 wave32s per WGP

typedef float v2f __attribute__((ext_vector_type(2)));
typedef float v8f __attribute__((ext_vector_type(8)));

// ---- monotonic float <-> uint32 key mapping --------------------------------
__device__ __forceinline__ unsigned sortkey(float f) {
  unsigned b = __float_as_uint(f);
  return b ^ ((b & 0x80000000u) ? 0xFFFFFFFFu : 0x80000000u);
}
__device__ __forceinline__ float unsortkey(unsigned u) {
  unsigned b = (u & 0x80000000u) ? (u ^ 0x80000000u) : ~u;
  return __uint_as_float(b);
}

// ---- inclusive prefix sum over 256 f32 held by ONE wave32 ------------------
// Element mapping matches the WMMA 32-bit C/D layout:
//   e = 16*(lane&15) + v + 8*(lane>>4),  v = VGPR index 0..7
// csum = L(16x16 lower-tri ones) x P  + per-column offsets.
// L is fed as the A operand (documented 16x4 f32 layout), P chunks as B.
__device__ __forceinline__ v8f wave_scan256(v8f p, float* smemP, float* smemT) {
  const int lane = threadIdx.x & 31;
  const int col  = lane & 15;
  const int half = lane >> 4;
#pragma unroll
  for (int v = 0; v < 8; ++v)
    smemP[16 * col + v + 8 * half] = p[v];      // element order in LDS
  // single-wave LDS: in-order; compiler inserts ds-cnt waits on use.
  v8f acc = {};
#pragma unroll
  for (int j = 0; j < 4; ++j) {                 // K chunks of 4
    v2f a, bm;
#pragma unroll
    for (int c = 0; c < 2; ++c) {
      const int K = 4 * j + c + 2 * half;       // documented A layout
      a[c]  = (K <= col) ? 1.0f : 0.0f;         // L[M=col][K]
      bm[c] = smemP[16 * col + K];              // P[K][N=col]
    }
    acc = __builtin_amdgcn_wmma_f32_16x16x4_f32(false, a, false, bm,
                                                (short)0, acc, false, false);
  }
  // column totals sit at M=15 => v=7 on lanes 16..31
  if (half == 1) smemT[col] = acc[7];
  float off = 0.0f;
#pragma unroll
  for (int n = 0; n < 15; ++n) off += (n < col) ? smemT[n] : 0.0f;
  v8f out;
#pragma unroll
  for (int v = 0; v < 8; ++v) out[v] = acc[v] + off;
  return out;
}

// ---- init: global running min of csum[:,0] (bits of a non-negative float) --
__global__ void k_init(unsigned* gmin) { *gmin = 0x7F800000u; }

// ---- kernel 1: per-row top-256 select + sort + softmax + WMMA cumsum -------
__global__ __launch_bounds__(THREADS) void k_topk(
    const float* __restrict__ logits, const float* __restrict__ params,
    float* __restrict__ wsVals, float* __restrict__ wsCsum,
    int* __restrict__ wsIdx, unsigned* __restrict__ gmin) {
  __shared__ unsigned hist[NBINS];              // 64 KB (320 KB LDS on CDNA5)
  __shared__ unsigned long long sbuf[512];      // 4 KB sort buffer
  __shared__ unsigned segsum[256];
  __shared__ unsigned s_thr, s_ctrGt, s_ctrEq;
  __shared__ float smemP[256];
  __shared__ float smemT[16];

  const int tid = threadIdx.x;
  const int row = blockIdx.x;
  const float4* __restrict__ rp4 =
      (const float4*)(logits + (size_t)row * VOCAB);

  for (int i = tid; i < NBINS; i += THREADS) hist[i] = 0u;
  if (tid == 0) { s_ctrGt = 0u; s_ctrEq = 0u; }
  sbuf[tid] = 0ull;
  __syncthreads();

  // Pass A: stream the row once as B128 loads; 14-bit LDS histogram
  for (int i = tid; i < VOCAB4; i += THREADS) {
    __builtin_prefetch((const float*)(rp4 + i + 2048), 0, 0);
    const float4 f = rp4[i];
    atomicAdd(&hist[sortkey(f.x) >> BSHIFT], 1u);
    atomicAdd(&hist[sortkey(f.y) >> BSHIFT], 1u);
    atomicAdd(&hist[sortkey(f.z) >> BSHIFT], 1u);
    atomicAdd(&hist[sortkey(f.w) >> BSHIFT], 1u);
  }
  __syncthreads();

  // parallel segment sums (threads 0..255), then serial threshold locate
  if (tid < 256) {
    unsigned ss = 0;
#pragma unroll 8
    for (int k2 = 0; k2 < SEG; ++k2) ss += hist[SEG * tid + k2];
    segsum[tid] = ss;
  }
  __syncthreads();
  if (tid == 0) {
    unsigned need = TOPK;
    int seg = 0;
    for (int s = 255; s >= 0; --s) {
      const unsigned c = segsum[s];
      if (c >= need) { seg = s; break; }
      need -= c;
    }
    int thr = seg * SEG;
    for (int b = seg * SEG + SEG - 1; b >= seg * SEG; --b) {
      const unsigned c = hist[b];
      if (c >= need) { thr = b; break; }
      need -= c;
    }
    s_thr = (unsigned)thr;
  }
  __syncthreads();
  const unsigned thr = s_thr;

  // Pass B: collect candidates (count(prefix>thr) < 256 by construction);
  // second stream mostly hits the 192 MB L2.
  for (int i = tid; i < VOCAB4; i += THREADS) {
    const float4 f = rp4[i];
    const float fl[4] = {f.x, f.y, f.z, f.w};
#pragma unroll
    for (int c = 0; c < 4; ++c) {
      const unsigned key = sortkey(fl[c]);
      const unsigned pre = key >> BSHIFT;
      if (pre >= thr) {
        const unsigned idx = (unsigned)(4 * i + c);
        // ~idx in low bits: descending u64 sort => key desc, ties by idx asc
        const unsigned long long pk =
            ((unsigned long long)key << 32) | (unsigned)(~idx);
        if (pre > thr) {
          const unsigned slot = atomicAdd(&s_ctrGt, 1u);
          if (slot < 256u) sbuf[slot] = pk;
        } else {
          const unsigned slot = atomicAdd(&s_ctrEq, 1u);
          if (slot < 256u) sbuf[256u + slot] = pk;
        }
      }
    }
  }
  __syncthreads();

  // bitonic sort, 512 elements descending, one element per thread
  for (unsigned k2 = 2; k2 <= 512; k2 <<= 1) {
    for (unsigned j = k2 >> 1; j > 0; j >>= 1) {
      const unsigned i = (unsigned)tid;
      const unsigned ixj = i ^ j;
      if (ixj > i) {
        const unsigned long long x = sbuf[i], y = sbuf[ixj];
        const bool sw = ((i & k2) == 0) ? (x < y) : (x > y);
        if (sw) { sbuf[i] = y; sbuf[ixj] = x; }
      }
      __syncthreads();
    }
  }

  // wave 0: mask + temperature + softmax + WMMA cumsum, spill to workspace
  if (tid < 32) {
    const int lane = tid, col = lane & 15, half = lane >> 4;
    const float tk = params[row * 3 + 0];
    const float tmp = params[row * 3 + 2];
    v8f x; int tok[8];
#pragma unroll
    for (int v = 0; v < 8; ++v) {
      const int e = 16 * col + v + 8 * half;
      const unsigned long long pk = sbuf[e];
      const float val = unsortkey((unsigned)(pk >> 32));
      tok[v] = (int)(~(unsigned)(pk & 0xFFFFFFFFull));
      const float m = ((float)e >= tk) ? IGNORED_V : val;  // ref masks THEN /T
      x[v] = m / tmp;
    }
    // softmax over the wave's 256 values
    float mx = x[0];
#pragma unroll
    for (int v = 1; v < 8; ++v) mx = fmaxf(mx, x[v]);
    for (int o = 16; o >= 1; o >>= 1) mx = fmaxf(mx, __shfl_xor(mx, o, 32));
    v8f p; float s = 0.0f;
#pragma unroll
    for (int v = 0; v < 8; ++v) { p[v] = __expf(x[v] - mx); s += p[v]; }
    for (int o = 16; o >= 1; o >>= 1) s += __shfl_xor(s, o, 32);
    const float inv = 1.0f / s;
#pragma unroll
    for (int v = 0; v < 8; ++v) p[v] *= inv;

    const v8f cs = wave_scan256(p, smemP, smemT);

    const size_t base = (size_t)row * 256;
#pragma unroll
    for (int v = 0; v < 8; ++v) {
      const int e = 16 * col + v + 8 * half;
      wsVals[base + e] = x[v];
      wsCsum[base + e] = cs[v];
      wsIdx[base + e] = tok[v];
    }
    if (lane == 0) atomicMin(gmin, __float_as_uint(cs[0]));  // csum[row][0]
  }
}

// ---- kernel 2: top-p mask + softmax + WMMA cumsum + sample -----------------
__global__ __launch_bounds__(32) void k_sample(const float* __restrict__ wsVals,
                                               const float* __restrict__ wsCsum,
                                               const int*   __restrict__ wsIdx,
                                               const float* __restrict__ params,
                                               const float* __restrict__ randu,
                                               const unsigned* __restrict__ gmin,
                                               int* __restrict__ out) {
  __shared__ float smemP[256];
  __shared__ float smemT[16];
  const int row = blockIdx.x;
  const int lane = threadIdx.x, col = lane & 15, half = lane >> 4;
  const float tpe = fmaxf(__uint_as_float(*gmin), params[row * 3 + 1]);
  const float ru = randu[row];
  const size_t base = (size_t)row * 256;

  v8f x;
#pragma unroll
  for (int v = 0; v < 8; ++v) {
    const int e = 16 * col + v + 8 * half;
    x[v] = (wsCsum[base + e] > tpe && e != 0) ? IGNORED_V : wsVals[base + e];
  }
  float mx = x[0];
#pragma unroll
  for (int v = 1; v < 8; ++v) mx = fmaxf(mx, x[v]);
  for (int o = 16; o >= 1; o >>= 1) mx = fmaxf(mx, __shfl_xor(mx, o, 32));
  v8f p; float s = 0.0f;
#pragma unroll
  for (int v = 0; v < 8; ++v) { p[v] = __expf(x[v] - mx); s += p[v]; }
  for (int o = 16; o >= 1; o >>= 1) s += __shfl_xor(s, o, 32);
  const float inv = 1.0f / s;
#pragma unroll
  for (int v = 0; v < 8; ++v) p[v] *= inv;

  const v8f cs2 = wave_scan256(p, smemP, smemT);

  int cnt = 0;
#pragma unroll
  for (int v = 0; v < 8; ++v) cnt += (ru > cs2[v]) ? 1 : 0;
  for (int o = 16; o >= 1; o >>= 1) cnt += __shfl_xor(cnt, o, 32);
  if (lane == 0) {
    if (cnt > 255) cnt = 255;                 // jax take_along_axis clamps
    out[row] = wsIdx[base + cnt];
  }
}

// ---- launcher --------------------------------------------------------------
extern "C" void kernel_launch(void* const* d_in, const int* in_sizes, int n_in,
                              void* d_out, int out_size, void* d_ws, size_t ws_size,
                              hipStream_t stream) {
  const float* logits = (const float*)d_in[0];   // (B, V) f32
  const float* params = (const float*)d_in[1];   // (B, 3) [top_k, top_p, temp]
  const float* randu  = (const float*)d_in[2];   // (B, 1) f32

  unsigned char* ws = (unsigned char*)d_ws;
  unsigned* gmin = (unsigned*)ws;                         // 4 B (+pad)
  float* wsVals  = (float*)(ws + 256);                    // B*256 f32
  float* wsCsum  = wsVals + (size_t)B_ROWS * 256;         // B*256 f32
  int*   wsIdx   = (int*)(wsCsum + (size_t)B_ROWS * 256); // B*256 i32
  int* out = (int*)d_out;

  k_init<<<1, 1, 0, stream>>>(gmin);
  k_topk<<<B_ROWS, THREADS, 0, stream>>>(logits, params, wsVals, wsCsum,
                                         wsIdx, gmin);
  k_sample<<<B_ROWS, 32, 0, stream>>>(wsVals, wsCsum, wsIdx, params, randu,
                                      gmin, out);
}